// GraphAttnLayer_20263655702745
// MI455X (gfx1250) — compile-verified
//
#include <hip/hip_runtime.h>

typedef __attribute__((ext_vector_type(2))) float v2f;
typedef __attribute__((ext_vector_type(8))) float v8f;

#define NN 1024
#define DD 128

// ---------------------------------------------------------------------------
// Kernel 1: GEMV via WMMA f32 16x16x4.
//   Y(1024x16) = inp(1024x128) * Wmat(128x16),
//   Wmat col0 = W[d]+W[d+128], col1 = W[d], col2 = W[d+128], cols 3..15 = 0.
//   ws[0..1023]=u, ws[1024..2047]=vA, ws[2048..3071]=vB.
// One wave (32 threads) per 16-row tile; 64 blocks x 32 threads.
// ---------------------------------------------------------------------------
__global__ void __launch_bounds__(32)
gemv_wmma_f32(const float* __restrict__ inp,
              const float* __restrict__ w,   // attn_weight flat, 256 floats
              float* __restrict__ ws) {
  const int lane  = threadIdx.x;        // 0..31
  const int tile  = blockIdx.x;         // 0..63
  const int col   = lane & 15;          // N index (B/C/D) and M row (A)
  const int khalf = (lane >> 4) * 2;    // 0 or 2: K sub-offset for this half-wave
  const int row   = tile * 16 + col;    // A-matrix row for this lane

  v8f c = {};
#pragma unroll
  for (int ks = 0; ks < DD; ks += 4) {
    const int k0 = ks + khalf;          // this lane's K, K+1
    v2f a, b;
    a.x = inp[row * DD + k0];
    a.y = inp[row * DD + k0 + 1];

    const float w0a = w[k0],       w0b = w[k0 + 128];
    const float w1a = w[k0 + 1],   w1b = w[k0 + 1 + 128];
    b.x = (col == 0) ? (w0a + w0b) : (col == 1) ? w0a : (col == 2) ? w0b : 0.0f;
    b.y = (col == 0) ? (w1a + w1b) : (col == 1) ? w1a : (col == 2) ? w1b : 0.0f;

    // D = A(16x4) * B(4x16) + C ; emits v_wmma_f32_16x16x4_f32
    c = __builtin_amdgcn_wmma_f32_16x16x4_f32(
        /*neg_a=*/false, a, /*neg_b=*/false, b,
        /*c_mod=*/(short)0, c, /*reuse_a=*/false, /*reuse_b=*/false);
  }

  // C/D layout: VGPR j, lanes 0-15 -> M=j, N=lane; lanes 16-31 -> M=j+8, N=lane-16
  if (col < 3) {
    const int half = lane >> 4;
#pragma unroll
    for (int j = 0; j < 8; ++j) {
      const int m = tile * 16 + j + half * 8;
      ws[col * NN + m] = c[j];
    }
  }
}

// ---------------------------------------------------------------------------
// Kernel 2: per-row mask -> leaky_relu -> softmax. 1024 blocks x 256 threads,
// 4 columns per thread (int4 adj load, float4 out store). Bandwidth-bound.
// ---------------------------------------------------------------------------
__global__ void __launch_bounds__(256)
masked_softmax_rows(const int* __restrict__ adj,
                    const float* __restrict__ ws,
                    float* __restrict__ out) {
  const int m = blockIdx.x;
  const int t = threadIdx.x;            // 0..255
  const float* __restrict__ u  = ws;
  const float* __restrict__ vA = ws + NN;
  const float* __restrict__ vB = ws + 2 * NN;

  const int n0 = t * 4;
  const int4 a4 = ((const int4*)(adj + (size_t)m * NN))[t];
  const int am[4] = {a4.x, a4.y, a4.z, a4.w};

  float pre[4];
  if (m < 512) {
    const float u0 = u[2 * m];
    const float u1 = u[2 * m + 1];
#pragma unroll
    for (int i = 0; i < 4; ++i) pre[i] = ((n0 + i) >= 512) ? u1 : u0;
  } else {
#pragma unroll
    for (int i = 0; i < 4; ++i) {
      const int r0 = (2 * (n0 + i)) & (NN - 1);   // even, so r0+1 in range
      pre[i] = vA[r0] + vB[r0 + 1];
    }
  }

  float val[4];
  float lmax = -3.402823466e38f;
#pragma unroll
  for (int i = 0; i < 4; ++i) {
    float v = (am[i] > 0) ? pre[i] : 0.0f;   // masked_fill(mask==0, 0)
    v = (v >= 0.0f) ? v : 0.01f * v;         // leaky_relu(0.01)
    val[i] = v;
    lmax = fmaxf(lmax, v);
  }

  __shared__ float red[8];
  const int wid = t >> 5;

  // wave32 max reduce
#pragma unroll
  for (int off = 16; off > 0; off >>= 1)
    lmax = fmaxf(lmax, __shfl_xor(lmax, off, 32));
  if ((t & 31) == 0) red[wid] = lmax;
  __syncthreads();
  float rowmax = red[0];
#pragma unroll
  for (int i = 1; i < 8; ++i) rowmax = fmaxf(rowmax, red[i]);
  __syncthreads();   // red reused below

  float e[4];
  float lsum = 0.0f;
#pragma unroll
  for (int i = 0; i < 4; ++i) {
    e[i] = __expf(val[i] - rowmax);
    lsum += e[i];
  }
#pragma unroll
  for (int off = 16; off > 0; off >>= 1)
    lsum += __shfl_xor(lsum, off, 32);
  if ((t & 31) == 0) red[wid] = lsum;
  __syncthreads();
  float rowsum = 0.0f;
#pragma unroll
  for (int i = 0; i < 8; ++i) rowsum += red[i];

  const float inv = 1.0f / rowsum;
  float4 o;
  o.x = e[0] * inv; o.y = e[1] * inv; o.z = e[2] * inv; o.w = e[3] * inv;
  ((float4*)(out + (size_t)m * NN))[t] = o;
}

// ---------------------------------------------------------------------------
extern "C" void kernel_launch(void* const* d_in, const int* in_sizes, int n_in,
                              void* d_out, int out_size, void* d_ws, size_t ws_size,
                              hipStream_t stream) {
  const float* inp = (const float*)d_in[0];   // (1,1024,128) f32
  const int*   adj = (const int*)d_in[1];     // (1,1024,1024) i32
  const float* aw  = (const float*)d_in[2];   // (8,32,1) f32 -> 256 flat
  float* out = (float*)d_out;                 // (1,1024,1024) f32
  float* ws  = (float*)d_ws;                  // needs 3*1024 floats = 12 KB

  gemv_wmma_f32<<<64, 32, 0, stream>>>(inp, aw, ws);
  masked_softmax_rows<<<NN, 256, 0, stream>>>(adj, ws, out);
}